// BicubicResize_9947144257962
// MI455X (gfx1250) — compile-verified
//
#include <hip/hip_runtime.h>

typedef float v2f __attribute__((ext_vector_type(2)));
typedef float v8f __attribute__((ext_vector_type(8)));

#define KMAX 96        // max densified K-span of a 16-row output block (<= 4*15+18 = 78)
#define KPAD 97        // LDS row stride (odd -> conflict-free across 64 banks)

constexpr int Bc  = 8, Cc = 3;
constexpr int Hc  = 1080, Wc = 1920;
constexpr int OHc = 270,  OWc = 480;
constexpr int nOB = (OHc + 15) / 16;          // 17 row-blocks
constexpr int WSPLIT = 4;                     // w-chunk groups per (bc, ob)
constexpr int NCHUNK = (Wc / 16) / WSPLIT;    // 30 chunks of 16 cols per group

// ---------------------------------------------------------------------------
// Pass 1 (H): T[bc, o, w] = sum_p w_h[o,p] * X[bc, idx_h[o,p], w]
// Densify the 16-row block's taps into a 16xK f32 tile in LDS, then
// T_tile(16x16) = Wd(16xK) @ X(Kx16) via V_WMMA_F32_16X16X4_F32 (exact f32).
// 256 threads = 8 waves; each wave owns independent 16-wide column chunks.
// ---------------------------------------------------------------------------
__global__ __launch_bounds__(256) void bicubic_pass1_wmma(
    const float* __restrict__ x, const float* __restrict__ w_h,
    const int* __restrict__ idx_h, float* __restrict__ t, int P) {
  int id = blockIdx.x;
  const int ws = id % WSPLIT; id /= WSPLIT;
  const int ob = id % nOB;    id /= nOB;
  const int bc = id;

  __shared__ float Wd[16][KPAD];
  __shared__ int s_min, s_max;

  const int tid = threadIdx.x;
  if (tid == 0) { s_min = 0x7fffffff; s_max = 0; }
  __syncthreads();

  const int o = ob * 16 + (tid & 15);
  if (tid < 16 && o < OHc) {
    int mn = 0x7fffffff, mx = 0;
    for (int p = 0; p < P; ++p) {
      const int idx = idx_h[o * P + p];
      mn = min(mn, idx);
      mx = max(mx, idx);
    }
    atomicMin(&s_min, mn);
    atomicMax(&s_max, mx);
  }
  for (int i = tid; i < 16 * KPAD; i += 256) (&Wd[0][0])[i] = 0.0f;
  __syncthreads();

  const int row_min = s_min;
  int span = s_max - row_min + 1;
  if (span > KMAX) span = KMAX;

  // Scatter taps (reflected indices may repeat within a row -> accumulate;
  // one thread per row, so no races).
  if (tid < 16 && o < OHc) {
    for (int p = 0; p < P; ++p) {
      const int k = idx_h[o * P + p] - row_min;
      if (k >= 0 && k < KMAX) Wd[tid][k] += w_h[o * P + p];
    }
  }
  __syncthreads();

  const int lane = tid & 31;
  const int lr   = lane & 15;   // M (A) / N (B,C) within half-wave
  const int hi   = lane >> 4;   // 0: rows K0/K1, 1: rows K2/K3
  const float* xplane = x + (long long)bc * Hc * Wc;
  float*       tplane = t + (long long)bc * OHc * Wc;
  const int nFull   = span >> 2;      // k-steps with all 4 rows in-bounds
  const bool hasTail = (span & 3) != 0;
  const bool fullBlk = (ob + 1) * 16 <= OHc;

  for (int wc = ws * NCHUNK + (tid >> 5); wc < (ws + 1) * NCHUNK; wc += 8) {
    const int w0 = wc << 4;
    const float* xcol = xplane + (long long)row_min * Wc + w0 + lr;
    const float* bp   = xcol + 2 * hi * Wc;
    v8f acc = {0.f, 0.f, 0.f, 0.f, 0.f, 0.f, 0.f, 0.f};

    #pragma unroll 2
    for (int i = 0; i < nFull; ++i) {
      const int kk = 4 * i + 2 * hi;
      v2f a, b;
      a.x = Wd[lr][kk];
      a.y = Wd[lr][kk + 1];
      b.x = bp[0];
      b.y = bp[Wc];
      bp += 4 * Wc;
      acc = __builtin_amdgcn_wmma_f32_16x16x4_f32(
          false, a, false, b, (short)0, acc, false, false);
    }
    if (hasTail) {                       // zero-padded Wd rows; clamp reads in-bounds
      const int kk = nFull * 4 + 2 * hi;
      v2f a, b;
      a.x = Wd[lr][kk];
      a.y = Wd[lr][kk + 1];
      const int rA = min(kk, span - 1);
      const int rB = min(kk + 1, span - 1);
      b.x = xcol[(long long)rA * Wc];
      b.y = xcol[(long long)rB * Wc];
      acc = __builtin_amdgcn_wmma_f32_16x16x4_f32(
          false, a, false, b, (short)0, acc, false, false);
    }

    // C/D layout: VGPR v -> M = v + 8*hi, N = lane&15
    float* sp = tplane + (long long)(ob * 16 + hi * 8) * Wc + w0 + lr;
    if (fullBlk) {
      #pragma unroll
      for (int v = 0; v < 8; ++v) sp[v * Wc] = acc[v];
    } else {
      #pragma unroll
      for (int v = 0; v < 8; ++v)
        if (ob * 16 + hi * 8 + v < OHc) sp[v * Wc] = acc[v];
    }
  }
}

// ---------------------------------------------------------------------------
// Pass 2 (W): out[bc, oh, ow] = sum_p w_w[ow,p] * T[bc, oh, idx_w[ow,p]]
// Stage one intermediate row (1920 f32, b128 loads) in LDS, gather + FMA.
// ---------------------------------------------------------------------------
__global__ __launch_bounds__(256) void bicubic_pass2(
    const float* __restrict__ t, const float* __restrict__ w_w,
    const int* __restrict__ idx_w, float* __restrict__ out, int P) {
  const int oh = blockIdx.x % OHc;
  const int bc = blockIdx.x / OHc;

  __shared__ float row[Wc];
  const float4* trow4 = (const float4*)(t + ((long long)bc * OHc + oh) * Wc);
  for (int i = threadIdx.x; i < Wc / 4; i += 256) ((float4*)row)[i] = trow4[i];
  __syncthreads();

  float* orow = out + ((long long)bc * OHc + oh) * OWc;
  for (int ow = threadIdx.x; ow < OWc; ow += 256) {
    float s = 0.0f;
    for (int p = 0; p < P; ++p) {
      const int c = idx_w[ow * P + p];
      s = fmaf(w_w[ow * P + p], row[c], s);
    }
    orow[ow] = s;
  }
}

extern "C" void kernel_launch(void* const* d_in, const int* in_sizes, int n_in,
                              void* d_out, int out_size, void* d_ws, size_t ws_size,
                              hipStream_t stream) {
  const float* x     = (const float*)d_in[0];
  const float* w_h   = (const float*)d_in[1];
  const int*   idx_h = (const int*)d_in[2];   // JAX x64-off: int64 -> int32 on device
  const float* w_w   = (const float*)d_in[3];
  const int*   idx_w = (const int*)d_in[4];

  const int P_h = in_sizes[1] / OHc;          // taps per output row (17 or 18)
  const int P_w = in_sizes[3] / OWc;

  float* t = (float*)d_ws;                    // intermediate: 24*270*1920*4 = 49.8 MB

  bicubic_pass1_wmma<<<dim3(Bc * Cc * nOB * WSPLIT), 256, 0, stream>>>(
      x, w_h, idx_h, t, P_h);
  bicubic_pass2<<<dim3(Bc * Cc * OHc), 256, 0, stream>>>(
      t, w_w, idx_w, (float*)d_out, P_w);
}